// RegistrationRecall_56831007261011
// MI455X (gfx1250) — compile-verified
//
#include <hip/hip_runtime.h>

// ---------------------------------------------------------------------------
// RegistrationRecall on MI455X (gfx1250, wave32)
//
// dist^2(i,j) = ||s_i||^2 + ||t_j||^2 - 2 s_i.t_j
// -2*s.t computed by V_WMMA_F32_16X16X4_F32 (A pre-scaled by -2, K padded to 4,
// C = inline 0). Target chunk staged into LDS once per block (TDM tensor_load
// when available), with B fragments + ||t||^2 precomputed once instead of
// per-wave. Inner loop: 2 WMMA + 2 ds_b64 + 2 ds_b32 + adds/min3 per 512 pairs.
// ---------------------------------------------------------------------------

typedef __attribute__((ext_vector_type(2))) float v2f;
typedef __attribute__((ext_vector_type(8))) float v8f;
typedef unsigned int u32x4 __attribute__((ext_vector_type(4)));
typedef int i32x8 __attribute__((ext_vector_type(8)));
typedef int i32x4 __attribute__((ext_vector_type(4)));

#define WAVES_PER_BLOCK  8
#define BLOCK_THREADS    (WAVES_PER_BLOCK * 32)
#define MAX_CHUNK_TILES  64            // 1024 target points per block chunk
#define FINALIZE_THREADS 256

#if defined(__AMDGCN__) && __has_builtin(__builtin_amdgcn_tensor_load_to_lds)
#define RR_HAVE_TDM 1
#else
#define RR_HAVE_TDM 0
#endif

// Kernel 0: reset per-source-point running-min array (uint bit-pattern of +inf).
__global__ void rr_init_kernel(unsigned int* __restrict__ mins, int n) {
    int i = blockIdx.x * blockDim.x + threadIdx.x;
    if (i < n) mins[i] = 0x7F800000u; // +inf bits; non-negative floats order as uints
}

// Kernel 1: block = 8 source 16-row tiles x one 1024-point target chunk.
__global__ __launch_bounds__(BLOCK_THREADS)
void rr_knn_wmma_kernel(const float* __restrict__ S, const float* __restrict__ T,
                        unsigned int* __restrict__ mins, int N, int M) {
    const int lane = threadIdx.x & 31;
    const int wave = threadIdx.x >> 5;
    const int half = lane >> 4;     // 0: lanes 0-15 (K0,K1), 1: lanes 16-31 (K2,K3)
    const int l16  = lane & 15;

    // B fragments (per tile, per lane) and ||t||^2, built once per block.
    __shared__ v2f   bfrag[MAX_CHUNK_TILES * 32];   // 16 KB
    __shared__ float tsqArr[MAX_CHUNK_TILES * 16];  //  4 KB

    // ---- This block's target chunk ----
    const int numTiles = M >> 4;
    const int tileBase = blockIdx.y * MAX_CHUNK_TILES;
    const int tiles    = (numTiles - tileBase < MAX_CHUNK_TILES) ? (numTiles - tileBase)
                                                                 : MAX_CHUNK_TILES;
    const int chunkPts = tiles * 16;
    const float* __restrict__ Tc = T + (size_t)tileBase * 48;  // 16 pts * 3 floats

#if RR_HAVE_TDM
    // Stage the raw chunk (<=12 KB contiguous) into LDS with the Tensor Data Mover.
    __shared__ float rawStage[MAX_CHUNK_TILES * 48];  // 12 KB
    if (wave == 0) {
        const unsigned lds_off =
            (unsigned)(unsigned long long)(__attribute__((address_space(3))) float*)rawStage;
        const unsigned long long gaddr = (unsigned long long)(const void*)Tc;
        const unsigned d0 = (unsigned)(chunkPts * 3);   // elements along dim0 (<=3072)

        u32x4 g0;
        g0[0] = 1u;                                        // count=1, user descriptor
        g0[1] = lds_off;                                   // LDS byte address
        g0[2] = (unsigned)(gaddr & 0xFFFFFFFFu);           // global_addr[31:0]
        g0[3] = (unsigned)((gaddr >> 32) & 0x01FFFFFFu)    // global_addr[56:32]
              | (2u << 30);                                // type = 2 ("image")
        i32x8 g1;
        g1[0] = (int)(2u << 16);                           // data_size = 4 bytes
        g1[1] = (int)((d0 & 0xFFFFu) << 16);               // tensor_dim0[15:0]
        g1[2] = (int)((d0 >> 16) | (1u << 16));            // tensor_dim0[31:16], dim1=1
        g1[3] = (int)((d0 & 0xFFFFu) << 16);               // tile_dim0 = d0
        g1[4] = 1;                                         // tile_dim1 = 1
        g1[5] = (int)d0;                                   // tensor_dim0_stride
        g1[6] = 0;
        g1[7] = 0;
        i32x4 z4 = {};
#if __clang_major__ >= 23
        i32x8 z8 = {};
        __builtin_amdgcn_tensor_load_to_lds(g0, g1, z4, z4, z8, 0);
#else
        __builtin_amdgcn_tensor_load_to_lds(g0, g1, z4, z4, 0);
#endif
        __builtin_amdgcn_s_wait_tensorcnt(0);
    }
    __syncthreads();
#define RR_RAW(i) rawStage[i]
#else
#define RR_RAW(i) Tc[i]
#endif

    // Prep: B fragments + ||t||^2, computed once per block (not once per wave).
    for (int p = threadIdx.x; p < chunkPts; p += BLOCK_THREADS) {
        const float x = RR_RAW(3 * p + 0);
        const float y = RR_RAW(3 * p + 1);
        const float z = RR_RAW(3 * p + 2);
        const int tile = p >> 4, l = p & 15;
        v2f lo; lo.x = x;  lo.y = y;     // lanes 0-15: {K0,K1}
        v2f hi; hi.x = z;  hi.y = 0.0f;  // lanes 16-31: {K2,K3}
        bfrag[tile * 32 + l]      = lo;
        bfrag[tile * 32 + 16 + l] = hi;
        tsqArr[p] = x * x + y * y + z * z;
    }
    __syncthreads();

    // ---- Source tile for this wave (clamped, keeps barriers/EXEC uniform) ----
    const int nSrcBlocks = N >> 4;
    int srcBlock = blockIdx.x * WAVES_PER_BLOCK + wave;
    if (srcBlock >= nSrcBlocks) srcBlock = nSrcBlocks - 1;
    const int srcBase = srcBlock * 16;

    const float sx = S[3 * (srcBase + l16) + 0];
    const float sy = S[3 * (srcBase + l16) + 1];
    const float sz = S[3 * (srcBase + l16) + 2];
    const float ssq = sx * sx + sy * sy + sz * sz;

    // A fragment (16x4 f32) pre-scaled by -2 so WMMA yields -2*s.t directly.
    v2f a;
    a.x = half ? (-2.0f * sz) : (-2.0f * sx);
    a.y = half ? 0.0f         : (-2.0f * sy);

    float runmin[8];
#pragma unroll
    for (int r = 0; r < 8; ++r) runmin[r] = __builtin_inff();

    const v8f cz = {};   // C = inline 0 in the WMMA encoding
    int t = 0;
    for (; t + 1 < tiles; t += 2) {
        const v2f  b0 = bfrag[t * 32 + lane];
        const v2f  b1 = bfrag[(t + 1) * 32 + lane];
        const float q0 = tsqArr[t * 16 + l16];
        const float q1 = tsqArr[(t + 1) * 16 + l16];

        const v8f acc0 = __builtin_amdgcn_wmma_f32_16x16x4_f32(
            false, a, false, b0, (short)0, cz, false, false);
        const v8f acc1 = __builtin_amdgcn_wmma_f32_16x16x4_f32(
            false, a, false, b1, (short)0, cz, false, false);

#pragma unroll
        for (int r = 0; r < 8; ++r)   // min(min(a,b),c) -> v_min3_num_f32
            runmin[r] = fminf(runmin[r], fminf(acc0[r] + q0, acc1[r] + q1));
    }
    if (t < tiles) {
        const v2f  b0 = bfrag[t * 32 + lane];
        const float q0 = tsqArr[t * 16 + l16];
        const v8f acc0 = __builtin_amdgcn_wmma_f32_16x16x4_f32(
            false, a, false, b0, (short)0, cz, false, false);
#pragma unroll
        for (int r = 0; r < 8; ++r)
            runmin[r] = fminf(runmin[r], acc0[r] + q0);
    }

    // ---- Min-reduce over the 16 column lanes inside each half (wave32) ----
#pragma unroll
    for (int r = 0; r < 8; ++r) {
        float v = runmin[r];
        v = fminf(v, __shfl_xor(v, 1));
        v = fminf(v, __shfl_xor(v, 2));
        v = fminf(v, __shfl_xor(v, 4));
        v = fminf(v, __shfl_xor(v, 8));
        runmin[r] = v;   // each half: chunk-min for row (r + 8*half)
    }

    // ---- Add ||s_row||^2, clamp, publish via uint atomicMin ----
#pragma unroll
    for (int r = 0; r < 8; ++r) {
        const int row = r + 8 * half;
        const float srow = __shfl(ssq, row);             // ssq lives in lanes 0-15
        const float d2 = fmaxf(srow + runmin[r], 0.0f);  // guard fp round-off
        if (l16 == 0)
            atomicMin(&mins[srcBase + row], __float_as_uint(d2));
    }
}

// Kernel 2: sum per-point min squared distances -> rmse -> success flag.
__global__ __launch_bounds__(FINALIZE_THREADS)
void rr_finalize_kernel(const unsigned int* __restrict__ mins, float* __restrict__ out, int N) {
    __shared__ float red[FINALIZE_THREADS];
    float s = 0.0f;
    for (int i = threadIdx.x; i < N; i += FINALIZE_THREADS)
        s += __uint_as_float(mins[i]);
    red[threadIdx.x] = s;
    __syncthreads();
    for (int off = FINALIZE_THREADS / 2; off > 0; off >>= 1) {
        if (threadIdx.x < off) red[threadIdx.x] += red[threadIdx.x + off];
        __syncthreads();
    }
    if (threadIdx.x == 0) {
        const float rmse = sqrtf(red[0] / (float)N);
        out[0] = (rmse < 0.1f) ? 1.0f : 0.0f;
    }
}

extern "C" void kernel_launch(void* const* d_in, const int* in_sizes, int n_in,
                              void* d_out, int out_size, void* d_ws, size_t ws_size,
                              hipStream_t stream) {
    const float* S = (const float*)d_in[0];   // source [N,3] f32
    const float* T = (const float*)d_in[1];   // target [M,3] f32
    const int N = in_sizes[0] / 3;
    const int M = in_sizes[1] / 3;

    unsigned int* mins = (unsigned int*)d_ws; // N uints: running min(dist^2) bits
    float* out = (float*)d_out;

    // 0) reset workspace every call (harness does not re-poison between replays)
    rr_init_kernel<<<(N + 255) / 256, 256, 0, stream>>>(mins, N);

    // 1) WMMA scan: grid.x = 128-source-row groups, grid.y = 64-tile target chunks
    const int srcGroups = (N / 16 + WAVES_PER_BLOCK - 1) / WAVES_PER_BLOCK;
    const int numTiles  = M / 16;
    const int chunks    = (numTiles + MAX_CHUNK_TILES - 1) / MAX_CHUNK_TILES;
    dim3 grid(srcGroups, chunks);
    rr_knn_wmma_kernel<<<grid, BLOCK_THREADS, 0, stream>>>(S, T, mins, N, M);

    // 2) reduce to the single success scalar
    rr_finalize_kernel<<<1, FINALIZE_THREADS, 0, stream>>>(mins, out, N);
}